// MoELayer_32375463478042
// MI455X (gfx1250) — compile-verified
//
#include <hip/hip_runtime.h>
#include <math.h>

#define B_DIM 32
#define S_DIM 2048
#define C_DIM 16
#define E_DIM 8
#define H_DIM 16
#define TWO_S 4096

typedef float v2f __attribute__((ext_vector_type(2)));
typedef float v8f __attribute__((ext_vector_type(8)));

__device__ __forceinline__ float softplus_f(float v) {
    // jax.nn.softplus == logaddexp(v, 0), numerically stable form
    return fmaxf(v, 0.0f) + log1pf(expf(-fabsf(v)));
}

// ---------------------------------------------------------------------------
// Kernel 1: gating.  One block (16 waves, 512 threads) per batch b.
// G[c][n], n in [0,16): n<8 -> xr·Wm[n], n>=8 -> xr·Wn[n-8], K = 4096.
// Each wave accumulates a partial 16x16 f32 tile over a K-slice of 256 via
// V_WMMA_F32_16X16X4_F32; partials are reduced through LDS; 16 lanes then do
// the top-k mask + softmax epilogue.
// ---------------------------------------------------------------------------
__global__ __launch_bounds__(512) void moe_gate_wmma(
    const float* __restrict__ x,
    const float* __restrict__ Wm, const float* __restrict__ bm,
    const float* __restrict__ Wn, const float* __restrict__ bn,
    const int* __restrict__ kptr,
    float* __restrict__ gate_out)
{
    __shared__ float pt[16][256];   // per-wave partial tiles (16 KB)
    __shared__ float gt[256];       // reduced tile

    const int b    = blockIdx.x;
    const int tid  = (int)threadIdx.x;
    const int wv   = tid >> 5;      // wave id 0..15 -> K slice
    const int lane = tid & 31;
    const int m    = lane & 15;     // A row (= c index) / B column
    const int half = lane >> 4;     // K sub-pair selector

    const float* xb = x + (size_t)b * (S_DIM * C_DIM * 2);
    // B column n: rows of Wm for n<8, rows of Wn for n>=8
    const float* wrow = (m < 8) ? (Wm + (size_t)m * TWO_S)
                                : (Wn + (size_t)(m - 8) * TWO_S);

    v8f acc0 = {0.f, 0.f, 0.f, 0.f, 0.f, 0.f, 0.f, 0.f};
    v8f acc1 = {0.f, 0.f, 0.f, 0.f, 0.f, 0.f, 0.f, 0.f};

    const int kbeg = wv * 256;
    const int kend = kbeg + 256;
    for (int k0 = kbeg; k0 < kend; k0 += 16) {
        // A fragment: lane holds A[c][j], A[c][j+1], j = k + 2*half,
        // A[c][j] = x[b, j/2, c, j%2]  -> contiguous float2 at s = k/2 + half
        v2f a0 = *(const v2f*)(xb + ((k0 >> 1) + 0 + half) * 32 + m * 2);
        v2f a1 = *(const v2f*)(xb + ((k0 >> 1) + 2 + half) * 32 + m * 2);
        v2f a2 = *(const v2f*)(xb + ((k0 >> 1) + 4 + half) * 32 + m * 2);
        v2f a3 = *(const v2f*)(xb + ((k0 >> 1) + 6 + half) * 32 + m * 2);
        // B fragment: lane holds B[2*half][n], B[2*half+1][n] = row_n[k+2*half..]
        v2f w0 = *(const v2f*)(wrow + k0 + 0  + 2 * half);
        v2f w1 = *(const v2f*)(wrow + k0 + 4  + 2 * half);
        v2f w2 = *(const v2f*)(wrow + k0 + 8  + 2 * half);
        v2f w3 = *(const v2f*)(wrow + k0 + 12 + 2 * half);
        acc0 = __builtin_amdgcn_wmma_f32_16x16x4_f32(false, a0, false, w0, (short)0, acc0, false, false);
        acc1 = __builtin_amdgcn_wmma_f32_16x16x4_f32(false, a1, false, w1, (short)0, acc1, false, false);
        acc0 = __builtin_amdgcn_wmma_f32_16x16x4_f32(false, a2, false, w2, (short)0, acc0, false, false);
        acc1 = __builtin_amdgcn_wmma_f32_16x16x4_f32(false, a3, false, w3, (short)0, acc1, false, false);
    }
    acc0 += acc1;

    // C layout: lane L holds column n=L%16, rows v + 8*(L/16)
#pragma unroll
    for (int v = 0; v < 8; ++v)
        pt[wv][(v + 8 * half) * 16 + m] = acc0[v];
    __syncthreads();

    // reduce 16 partial tiles
    if (tid < 256) {
        float s = 0.f;
#pragma unroll
        for (int w = 0; w < 16; ++w) s += pt[w][tid];
        gt[tid] = s;
    }
    __syncthreads();

    if (tid < 16) {
        const int kv = *kptr;
        float g[E_DIM];
#pragma unroll
        for (int e = 0; e < E_DIM; ++e) {
            float gm = gt[tid * 16 + e]     + bm[e];
            float gn = gt[tid * 16 + 8 + e] + bn[e];
            g[e] = gm + softplus_f(gn);
        }
        // full sort (8 elements) to get kth = sorted[E - kv - 1]
        float srt[E_DIM];
#pragma unroll
        for (int e = 0; e < E_DIM; ++e) srt[e] = g[e];
#pragma unroll
        for (int i = 0; i < E_DIM - 1; ++i)
#pragma unroll
            for (int j = 0; j < E_DIM - 1 - i; ++j)
                if (srt[j + 1] < srt[j]) { float t = srt[j]; srt[j] = srt[j + 1]; srt[j + 1] = t; }
        float kth = srt[E_DIM - kv - 1];
        float mx  = srt[E_DIM - 1];
        float w[E_DIM];
        float denom = 0.f;
#pragma unroll
        for (int e = 0; e < E_DIM; ++e) {
            float ev = (g[e] > kth) ? expf(g[e] - mx) : 0.0f;
            w[e] = ev;
            denom += ev;
        }
        float inv = 1.0f / denom;
        float* dst = gate_out + ((size_t)b * 16 + tid) * E_DIM;
#pragma unroll
        for (int e = 0; e < E_DIM; ++e) dst[e] = w[e] * inv;
    }
}

// ---------------------------------------------------------------------------
// Kernel 2: experts + combine.
// Block (u, b): covers s2 in [256u, 256u+256), all c2, so output is one
// contiguous 32KB slab. Inputs are 16 contiguous 2KB slabs staged via LDS.
// Loop iteration i == c2, so gate weights are block-uniform -> skip the
// (E-k)=6 experts whose gate weight is exactly zero.
// ---------------------------------------------------------------------------
__device__ __forceinline__ float apply_nlin(int e, float v) {
    switch (e) {
        case 0: return fmaxf(v, 0.0f);                                   // relu
        case 1: return tanhf(v);                                         // tanh
        case 2: return v > 0.0f ? v : expm1f(v);                         // elu
        case 3: return v / (1.0f + expf(-v));                            // silu
        case 4: return v;                                                // none
        case 5: return 0.5f * v * (1.0f + erff(v * 0.7071067811865475f));// gelu exact
        case 6: {                                                        // selu
            const float sc = 1.0507009873554805f, al = 1.6732632423543772f;
            return v > 0.0f ? sc * v : sc * al * expm1f(v);
        }
        default: return softplus_f(v);                                   // softplus
    }
}

__global__ __launch_bounds__(256) void moe_expert(
    const float* __restrict__ x,
    const float* __restrict__ W1, const float* __restrict__ b1,
    const float* __restrict__ W2, const float* __restrict__ b2,
    const float* __restrict__ gate,
    float* __restrict__ out)
{
    __shared__ __align__(16) float xs[16 * 512];          // input slabs  (32 KB)
    __shared__ float ys[256 * 33];                        // padded out tile
    __shared__ float w1s[E_DIM * H_DIM * 3];
    __shared__ float b1s[E_DIM * H_DIM];
    __shared__ float w2s[E_DIM * 2 * H_DIM];
    __shared__ float b2s[E_DIM * 2];
    __shared__ float gts[16 * E_DIM];

    const int u   = blockIdx.x;   // 0..7  : s2-range selector
    const int b   = blockIdx.y;   // 0..31
    const int tid = (int)threadIdx.x;

    if (tid < 384) w1s[tid] = W1[tid];
    if (tid < 128) b1s[tid] = b1[tid];
    if (tid < 256) w2s[tid] = W2[tid];
    if (tid < 16)  b2s[tid] = b2[tid];
    if (tid < 128) gts[tid] = gate[(size_t)b * 128 + tid];

    // stage 16 chunks: chunk c2 = x[b, 128*c2+16u : +16, :, :] (512 floats)
    const float4* xv = (const float4*)x;
#pragma unroll
    for (int i = 0; i < 8; ++i) {
        int f4 = tid + 256 * i;          // 0..2047 float4s
        int c2 = f4 >> 7;
        int w  = f4 & 127;
        size_t gbase = (size_t)(b * 2048 + 128 * c2 + 16 * u) * 8;  // in float4
        ((float4*)xs)[c2 * 128 + w] = xv[gbase + w];
    }
    __syncthreads();

#pragma unroll 1
    for (int i = 0; i < 16; ++i) {       // i == c2 (uniform per iteration)
        const int c2 = i;
        const int rc = tid;              // local s2 = rc, c = rc%16, j = rc/16
        float x0  = xs[c2 * 512 + rc * 2 + 0];
        float x1  = xs[c2 * 512 + rc * 2 + 1];
        float amp = sqrtf(x0 * x0 + x1 * x1);
        float acc0 = 0.f, acc1 = 0.f;
#pragma unroll
        for (int e = 0; e < E_DIM; ++e) {
            float ge = gts[c2 * 8 + e];
            if (ge != 0.0f) {            // block-uniform branch: skip dead experts
                float h0 = 0.f, h1 = 0.f;
#pragma unroll
                for (int h = 0; h < H_DIM; ++h) {
                    const float* wp = &w1s[(e * 16 + h) * 3];
                    float pre = fmaf(wp[0], x0,
                                fmaf(wp[1], x1,
                                fmaf(wp[2], amp, b1s[e * 16 + h])));
                    float hv = apply_nlin(e, pre);
                    h0 = fmaf(w2s[(e * 2 + 0) * 16 + h], hv, h0);
                    h1 = fmaf(w2s[(e * 2 + 1) * 16 + h], hv, h1);
                }
                acc0 = fmaf(ge, h0 + b2s[e * 2 + 0], acc0);
                acc1 = fmaf(ge, h1 + b2s[e * 2 + 1], acc1);
            }
        }
        // output element (s2=rc, c2, t) -> padded tile row rc
        ys[rc * 33 + c2 * 2 + 0] = acc0;
        ys[rc * 33 + c2 * 2 + 1] = acc1;
    }
    __syncthreads();

    // coalesced writeback: 8192 contiguous floats at out[b, 256u : 256u+256, :, :]
    float* outbase = out + (size_t)(b * 2048 + 256 * u) * 32;
#pragma unroll
    for (int i = 0; i < 32; ++i) {
        int f = tid + 256 * i;           // 0..8191
        outbase[f] = ys[(f >> 5) * 33 + (f & 31)];
    }
}

extern "C" void kernel_launch(void* const* d_in, const int* in_sizes, int n_in,
                              void* d_out, int out_size, void* d_ws, size_t ws_size,
                              hipStream_t stream) {
    const float* x  = (const float*)d_in[0];
    const float* Wm = (const float*)d_in[1];
    const float* bm = (const float*)d_in[2];
    const float* Wn = (const float*)d_in[3];
    const float* bn = (const float*)d_in[4];
    const float* W1 = (const float*)d_in[5];
    const float* b1 = (const float*)d_in[6];
    const float* W2 = (const float*)d_in[7];
    const float* b2 = (const float*)d_in[8];
    const int*   kp = (const int*)d_in[9];
    float* gate_ws  = (float*)d_ws;      // 512 * 8 floats = 16 KB

    moe_gate_wmma<<<dim3(B_DIM), dim3(512), 0, stream>>>(x, Wm, bm, Wn, bn, kp, gate_ws);
    moe_expert<<<dim3(8, B_DIM), dim3(256), 0, stream>>>(x, W1, b1, W2, b2, gate_ws,
                                                         (float*)d_out);
}